// TruncationMapper_67388036874244
// MI455X (gfx1250) — compile-verified
//
#include <hip/hip_runtime.h>

// Problem constants from the reference:
//   B=2, T=2, ENS=1, G=262144, GT=40962, F=64
//   x: (B, T, ENS, G, F) f32; we use the t = T-1 slice:
//     batch b slice base = ((b*T + (T-1))*ENS)*G*F = (2b+1)*G*F
//   out: (B, ENS, G, F) f32 flat = b*G*F + g*F + f
static constexpr long kG  = 262144;
static constexpr long kGT = 40962;
static constexpr long kF  = 64;

// ---------------------------------------------------------------------------
// Grid-stride zero fill with 128-bit stores (all regions are multiples of
// 4 floats). Harness poisons d_out/d_ws with 0xAA, so we re-zero every call.
__global__ void zero_f32x4(float4* __restrict__ p, int n4) {
    int i = blockIdx.x * blockDim.x + threadIdx.x;
    const int stride = gridDim.x * blockDim.x;
    const float4 z = make_float4(0.f, 0.f, 0.f, 0.f);
    for (; i < n4; i += stride) p[i] = z;
}

// ---------------------------------------------------------------------------
// norm[dst] += exp(-0.5*len^2)  (segment-sum of raw weights per destination)
__global__ void norm_accum(const int* __restrict__ dstIdx,
                           const float* __restrict__ elen,
                           float* __restrict__ norm, int E) {
    int e = blockIdx.x * blockDim.x + threadIdx.x;
    if (e >= E) return;
    float l = elen[e];
    atomicAdd(&norm[dstIdx[e]], __expf(-0.5f * l * l));
}

// ---------------------------------------------------------------------------
// gfx1250 async global->LDS gather (ASYNCcnt-tracked data path).
// VDST holds the per-lane LDS byte address, VADDR the 64-bit global address.
__device__ __forceinline__ void async_gather_b128(unsigned lds_addr, const void* gaddr) {
    asm volatile("global_load_async_to_lds_b128 %0, %1, off"
                 :: "v"(lds_addr), "v"(gaddr) : "memory");
}
__device__ __forceinline__ void wait_async0() {
    asm volatile("s_wait_asynccnt 0" ::: "memory");
}

// ---------------------------------------------------------------------------
// One projection hop:  y[b][dst][:] += w_e * x[b][src][:]   for both batches.
// Layout: 16 lanes per edge (float4 each -> 64 feats), 16 edges per 256-thread
// block (8 wave32s). Source rows are staged through the async global->LDS
// engine; the per-edge weight (exp + normalizer gather) is computed while the
// async gather is in flight; results scatter via f32 global atomics whose
// accumulation targets live in the 192 MB L2.
__global__ void proj_scatter(const int* __restrict__ srcIdx,
                             const int* __restrict__ dstIdx,
                             const float* __restrict__ elen,
                             const float* __restrict__ norm,
                             const float* __restrict__ x0,  // batch 0 source base
                             const float* __restrict__ x1,  // batch 1 source base
                             float* __restrict__ y0,        // batch 0 dest base
                             float* __restrict__ y1,        // batch 1 dest base
                             int E) {
    __shared__ float4 stage[512];                 // 256 lanes x 2 batches x 16B
    const int tid   = threadIdx.x;
    const int q     = tid & 15;                   // feature quad 0..15
    const int eslot = tid >> 4;                   // edge slot within block
    const long e = (long)blockIdx.x * 16 + eslot;
    if (e >= E) return;

    const int src = srcIdx[e];
    const int dst = dstIdx[e];
    const float l = elen[e];

    // Per-lane LDS byte addresses (low 32 bits of the generic shared pointer
    // are the LDS offset on gfx1250).
    const unsigned lds0 = (unsigned)(unsigned long long)&stage[tid];
    const unsigned lds1 = (unsigned)(unsigned long long)&stage[256 + tid];
    const float4* g0 = (const float4*)(x0 + (size_t)src * kF) + q;
    const float4* g1 = (const float4*)(x1 + (size_t)src * kF) + q;

    // Kick off both row gathers first ...
    async_gather_b128(lds0, g0);
    async_gather_b128(lds1, g1);

    // ... and overlap the weight computation (normalizer gather + v_exp_f32)
    // with the async engine. The "memory" clobbers keep this between issue
    // and wait.
    const float w = __expf(-0.5f * l * l) / (norm[dst] + 1e-8f);

    wait_async0();

    const float4 v0 = stage[tid];
    const float4 v1 = stage[256 + tid];

    float* o0 = y0 + (size_t)dst * kF + q * 4;
    float* o1 = y1 + (size_t)dst * kF + q * 4;
    atomicAdd(o0 + 0, v0.x * w);
    atomicAdd(o0 + 1, v0.y * w);
    atomicAdd(o0 + 2, v0.z * w);
    atomicAdd(o0 + 3, v0.w * w);
    atomicAdd(o1 + 0, v1.x * w);
    atomicAdd(o1 + 1, v1.y * w);
    atomicAdd(o1 + 2, v1.z * w);
    atomicAdd(o1 + 3, v1.w * w);
}

// ---------------------------------------------------------------------------
extern "C" void kernel_launch(void* const* d_in, const int* in_sizes, int n_in,
                              void* d_out, int out_size, void* d_ws, size_t ws_size,
                              hipStream_t stream) {
    const float* x        = (const float*)d_in[0];
    const int*   ei_down  = (const int*)  d_in[1];   // (2, E_DOWN): row0=src, row1=dst
    const float* el_down  = (const float*)d_in[2];
    const int*   ei_up    = (const int*)  d_in[3];   // (2, E_UP)
    const float* el_up    = (const float*)d_in[4];

    const int E_down = in_sizes[1] / 2;
    const int E_up   = in_sizes[3] / 2;

    float* out = (float*)d_out;

    // Workspace carve-up (floats): xt[2*GT*F] | norm_down[GT pad64] | norm_up[G]
    // Total ~22.3 MB. All region sizes are multiples of 4 floats.
    float* xt        = (float*)d_ws;
    float* norm_down = xt + 2 * kGT * kF;
    float* norm_up   = norm_down + ((kGT + 63) & ~63L);
    const long ws_floats = 2 * kGT * kF + ((kGT + 63) & ~63L) + kG;

    // 1) Zero the accumulation targets (128-bit stores).
    zero_f32x4<<<1024, 256, 0, stream>>>((float4*)xt, (int)(ws_floats / 4));
    zero_f32x4<<<4096, 256, 0, stream>>>((float4*)out, (int)(2L * kG * kF / 4));

    // 2) Per-destination weight normalizers.
    norm_accum<<<(E_down + 255) / 256, 256, 0, stream>>>(ei_down + E_down, el_down,
                                                         norm_down, E_down);
    norm_accum<<<(E_up + 255) / 256, 256, 0, stream>>>(ei_up + E_up, el_up,
                                                       norm_up, E_up);

    // 3) Down projection: x[:, T-1] (G-space) -> xt (GT-space).
    const float* xb0 = x + 1L * kG * kF;   // (b=0, t=1)
    const float* xb1 = x + 3L * kG * kF;   // (b=1, t=1)
    proj_scatter<<<(E_down + 15) / 16, 256, 0, stream>>>(
        ei_down, ei_down + E_down, el_down, norm_down,
        xb0, xb1, xt, xt + kGT * kF, E_down);

    // 4) Up projection: xt (GT-space) -> out (G-space).
    proj_scatter<<<(E_up + 15) / 16, 256, 0, stream>>>(
        ei_up, ei_up + E_up, el_up, norm_up,
        xt, xt + kGT * kF, out, out + kG * kF, E_up);
}